// DynamicRoIAlign_52913997087012
// MI455X (gfx1250) — compile-verified
//
#include <hip/hip_runtime.h>
#include <stdint.h>

typedef unsigned int v4u __attribute__((ext_vector_type(4)));
typedef int          v8i __attribute__((ext_vector_type(8)));
typedef int          v4i __attribute__((ext_vector_type(4)));

#define HH   160
#define WW   160
#define CC   256
#define OHH  14
#define OWW  14
#define NPOS (OHH * OWW)   /* 196 */
#define HWSZ (HH * WW)     /* 25600 */

__device__ __forceinline__ int iclamp(int v, int lo, int hi) {
    return v < lo ? lo : (v > hi ? hi : v);
}

__global__ __launch_bounds__(256) void roi_align_gfx1250_kernel(
    const float* __restrict__ feat,   // (8,256,160,160)
    const float* __restrict__ rois,   // (1024,5)
    float* __restrict__ out)          // (1024,256,14,14)
{
    __shared__ __align__(16) float s_roi[8];
    __shared__ __align__(16) int   s_off[NPOS][4];
    __shared__ __align__(16) float s_w[NPOS][4];

    const unsigned k   = blockIdx.x;
    const unsigned tid = threadIdx.x;

    // ---- CDNA5 Tensor Data Mover: DMA this ROI's 5 floats into LDS ----
    {
        uint64_t ga  = (uint64_t)(uintptr_t)(rois + (size_t)k * 5u);
        uint32_t lao = (uint32_t)(uintptr_t)(&s_roi[0]);   // flat low 32 bits == LDS offset

        v4u g0;
        g0[0] = 1u;                                        // count=1 valid descriptor
        g0[1] = lao;                                       // lds_addr (bytes)
        g0[2] = (uint32_t)ga;                              // global_addr[31:0]
        g0[3] = ((uint32_t)(ga >> 32) & 0x01FFFFFFu)       // global_addr[56:32]
              | (2u << 30);                                // type = 2 ("image")

        v8i g1;
        g1[0] = (int)(2u << 16);   // wg_mask=0, data_size=2 (4 bytes)
        g1[1] = (int)(5u << 16);   // tensor_dim0 = 5 elements
        g1[2] = (int)(1u << 16);   // tensor_dim1 = 1
        g1[3] = (int)(5u << 16);   // tile_dim0 = 5
        g1[4] = 1;                 // tile_dim1 = 1, tile_dim2 = 0 (unused)
        g1[5] = 5;                 // tensor_dim0_stride = 5
        g1[6] = (int)(5u << 16);   // tensor_dim1_stride = 5 (1-row tile; benign)
        g1[7] = 0;

        v4i g2 = {0, 0, 0, 0};     // dims 2..4 unused
        v4i g3 = {0, 0, 0, 0};
        v8i g4 = {0, 0, 0, 0, 0, 0, 0, 0};   // trailing group storage (clang-23 form)

        __builtin_amdgcn_tensor_load_to_lds(g0, g1, g2, g3, g4, 0);
        __builtin_amdgcn_s_wait_tensorcnt(0);
    }

    const int   b  = iclamp((int)s_roi[0], 0, 7);
    const float x1 = s_roi[1] * 160.0f;
    const float y1 = s_roi[2] * 160.0f;
    const float x2 = s_roi[3] * 160.0f;
    const float y2 = s_roi[4] * 160.0f;

    // ---- Phase 1: per-position offsets + fused weights (channel-invariant) ----
    if (tid < NPOS) {
        const unsigned oh = tid / OWW;
        const unsigned ow = tid - oh * OWW;

        const float fx  = x1 + (float)ow * (1.0f / 13.0f) * (x2 - x1);
        const float fy  = y1 + (float)oh * (1.0f / 13.0f) * (y2 - y1);
        const float ixf = fx - 0.5f;             // unaligned grid-sample math
        const float iyf = fy - 0.5f;

        const float xfl = floorf(ixf), yfl = floorf(iyf);
        const int   x0  = (int)xfl,    y0  = (int)yfl;
        const int   x1i = x0 + 1,      y1i = y0 + 1;
        const float wx1 = ixf - xfl,   wy1 = iyf - yfl;
        const float wx0 = 1.0f - wx1,  wy0 = 1.0f - wy1;

        const float vx0 = (x0  >= 0 && x0  < WW) ? 1.0f : 0.0f;
        const float vx1 = (x1i >= 0 && x1i < WW) ? 1.0f : 0.0f;
        const float vy0 = (y0  >= 0 && y0  < HH) ? 1.0f : 0.0f;
        const float vy1 = (y1i >= 0 && y1i < HH) ? 1.0f : 0.0f;

        const int x0c = iclamp(x0,  0, WW - 1);
        const int x1c = iclamp(x1i, 0, WW - 1);
        const int y0c = iclamp(y0,  0, HH - 1);
        const int y1c = iclamp(y1i, 0, HH - 1);

        s_off[tid][0] = y0c * WW + x0c;
        s_off[tid][1] = y0c * WW + x1c;
        s_off[tid][2] = y1c * WW + x0c;
        s_off[tid][3] = y1c * WW + x1c;

        s_w[tid][0] = wy0 * wx0 * vy0 * vx0;
        s_w[tid][1] = wy0 * wx1 * vy0 * vx1;
        s_w[tid][2] = wy1 * wx0 * vy1 * vx0;
        s_w[tid][3] = wy1 * wx1 * vy1 * vx1;
    }
    __syncthreads();

    // ---- Phase 2: 256*196 outputs; coalesced NT stores; division-free (c,p) walk ----
    const float* __restrict__ fb = feat + (size_t)b * ((size_t)CC * HWSZ);
    float*       __restrict__ ob = out  + (size_t)k * ((size_t)CC * NPOS);

    // e = rep*256 + tid  ==>  start (c,p) = (tid/196, tid%196); step: c+=1, p+=60, wrap.
    unsigned c = (tid >= (unsigned)NPOS) ? 1u : 0u;
    unsigned p = tid - c * (unsigned)NPOS;

    for (unsigned rep = 0; rep < NPOS; ++rep) {
        const float* __restrict__ fc = fb + (size_t)c * HWSZ;

        const int4   o = *reinterpret_cast<const int4*>(&s_off[p][0]);   // ds_load_b128
        const float4 w = *reinterpret_cast<const float4*>(&s_w[p][0]);   // ds_load_b128

        float acc = fc[o.x] * w.x;
        acc = fmaf(fc[o.y], w.y, acc);
        acc = fmaf(fc[o.z], w.z, acc);
        acc = fmaf(fc[o.w], w.w, acc);

        // Prefetch next channel's matching corner (gfx1250 global_prefetch_b8):
        // next rep advances c by 1 (or 2) for this lane; feat is L2-resident, this
        // hides L2->L0 latency in a latency-bound gather loop.
        __builtin_prefetch(fc + HWSZ + o.x, 0, 0);

        // Streaming output: non-temporal store so the 205MB write-once output
        // does not evict the L2-resident feature map (192MB L2 vs 210MB feat).
        __builtin_nontemporal_store(acc, ob + c * (unsigned)NPOS + p);

        // advance e by 256 = 196 + 60 without a division
        c += 1u;
        p += 60u;
        if (p >= (unsigned)NPOS) { p -= (unsigned)NPOS; c += 1u; }
    }
}

extern "C" void kernel_launch(void* const* d_in, const int* in_sizes, int n_in,
                              void* d_out, int out_size, void* d_ws, size_t ws_size,
                              hipStream_t stream) {
    const float* feat = (const float*)d_in[0];
    const float* rois = (const float*)d_in[1];
    float* out = (float*)d_out;

    const int K = in_sizes[1] / 5;   // 1024 ROIs
    roi_align_gfx1250_kernel<<<dim3(K), dim3(256), 0, stream>>>(feat, rois, out);
}